// GroupSort_4750233829451
// MI455X (gfx1250) — compile-verified
//
#include <hip/hip_runtime.h>

// GroupSort-2 over channel pairs:
//   out[n,2c]   = min(x[n,2c], x[n,2c+1])
//   out[n,2c+1] = max(x[n,2c], x[n,2c+1])
// x shape (32, 256, 56, 56) f32.  Pure streaming op: 205.5 MB traffic,
// ~8.8 us floor at 23.3 TB/s.  Strategy: b128 NT loads/stores, wave32,
// one float4 per plane per lane.

typedef __attribute__((ext_vector_type(4))) float v4f;

__global__ __launch_bounds__(256) void groupsort_pair_minmax(
    const float* __restrict__ x, float* __restrict__ out, int nthreads) {
  const int i = blockIdx.x * blockDim.x + threadIdx.x;
  if (i >= nthreads) return;

  constexpr int VEC_PER_PLANE = (56 * 56) / 4;  // 784 float4 per HxW plane

  // i -> (channel-pair, vec4-within-plane); div-by-const = umulhi+shift
  const int pair = i / VEC_PER_PLANE;
  const int v    = i - pair * VEC_PER_PLANE;

  const int e0 = pair * (2 * VEC_PER_PLANE) + v;  // even channel, vec4 units
  const int e1 = e0 + VEC_PER_PLANE;              // odd channel (+3136 floats)

  const v4f* __restrict__ xv = (const v4f*)x;
  v4f* __restrict__ ov       = (v4f*)out;

  // Streamed-once data: non-temporal loads/stores (gfx1250 th:TH_*_NT),
  // 192 MB L2 is smaller than the 205 MB working set anyway.
  v4f a = __builtin_nontemporal_load(xv + e0);
  v4f b = __builtin_nontemporal_load(xv + e1);

  v4f lo, hi;
  lo.x = __builtin_fminf(a.x, b.x);  hi.x = __builtin_fmaxf(a.x, b.x);
  lo.y = __builtin_fminf(a.y, b.y);  hi.y = __builtin_fmaxf(a.y, b.y);
  lo.z = __builtin_fminf(a.z, b.z);  hi.z = __builtin_fmaxf(a.z, b.z);
  lo.w = __builtin_fminf(a.w, b.w);  hi.w = __builtin_fmaxf(a.w, b.w);

  __builtin_nontemporal_store(lo, ov + e0);
  __builtin_nontemporal_store(hi, ov + e1);
}

extern "C" void kernel_launch(void* const* d_in, const int* in_sizes, int n_in,
                              void* d_out, int out_size, void* d_ws, size_t ws_size,
                              hipStream_t stream) {
  (void)n_in; (void)out_size; (void)d_ws; (void)ws_size;

  const float* x = (const float*)d_in[0];
  float* out     = (float*)d_out;

  const int n_elem  = in_sizes[0];          // 32*256*56*56 = 25,690,112
  const int hw      = 56 * 56;              // 3136
  const int pairs   = n_elem / (2 * hw);    // 32*128 = 4096 channel pairs
  const int vecs    = hw / 4;               // 784 float4 per plane
  const int threads = pairs * vecs;         // 3,211,264 lanes (8 floats each)

  const int block = 256;                    // 8 wave32s per block
  const int grid  = (threads + block - 1) / block;  // 12,544 blocks

  groupsort_pair_minmax<<<grid, block, 0, stream>>>(x, out, threads);
}